// GNNGroupTracker_63282048139481
// MI455X (gfx1250) — compile-verified
//
#include <hip/hip_runtime.h>
#include <math.h>

#define GN 20000      // nodes
#define GE 160000     // edges
#define GEN (GE + GN) // edges + self loops
#define GD 64
#define GHC 256       // H*C

typedef float v2f __attribute__((ext_vector_type(2)));
typedef float v8f __attribute__((ext_vector_type(8)));

#define CDIV(a, b) (((a) + (b) - 1) / (b))

// ---------- float <-> orderable-uint (for atomic max on f32) ----------
__device__ __forceinline__ unsigned f2ord(float f) {
    unsigned u = __float_as_uint(f);
    return (u & 0x80000000u) ? ~u : (u | 0x80000000u);
}
__device__ __forceinline__ float ord2f(unsigned u) {
    return (u & 0x80000000u) ? __uint_as_float(u & 0x7fffffffu)
                             : __uint_as_float(~u);
}

// ---------- WMMA f32 GEMM: Y[M,Ncol] = act(X[M,K] @ W[K,Ncol] + bias) ----------
// K compile-time (fully unrolled WMMA clause). One wave32 per 16x32 output
// strip: two accumulators share one A fragment (halves A traffic, doubles
// WMMA density). M mult of 16, Ncol mult of 32.
template <int K>
__global__ void k_gemm_wmma(const float* __restrict__ X, const float* __restrict__ W,
                            const float* __restrict__ bias, float* __restrict__ Y,
                            int M, int Ncol, int act) {
    int wave = (blockIdx.x * blockDim.x + threadIdx.x) >> 5;
    int lane = threadIdx.x & 31;
    int ntN2 = Ncol >> 5;                 // 16x32 strips per row-panel
    int ntM = M >> 4;
    if (wave >= ntM * ntN2) return;
    int tm = wave / ntN2;
    int tn2 = wave - tm * ntN2;
    int half = lane >> 4;                 // 0: lanes 0-15, 1: lanes 16-31
    int l16 = lane & 15;

    const float* Arow = X + (size_t)(tm * 16 + l16) * K;   // A row M=l16
    const float* B0 = W + (size_t)(tn2 * 32 + l16);        // B col N=l16
    const float* B1 = B0 + 16;
    v8f acc0 = {}, acc1 = {};
#pragma unroll
    for (int k = 0; k < K; k += 4) {
        int ka = k + 2 * half;                             // K split across lane halves
        v2f a;  a.x  = Arow[ka];               a.y  = Arow[ka + 1];
        v2f b0; b0.x = B0[(size_t)ka * Ncol];  b0.y = B0[(size_t)(ka + 1) * Ncol];
        v2f b1; b1.x = B1[(size_t)ka * Ncol];  b1.y = B1[(size_t)(ka + 1) * Ncol];
        acc0 = __builtin_amdgcn_wmma_f32_16x16x4_f32(false, a, false, b0,
                                                     (short)0, acc0, false, false);
        acc1 = __builtin_amdgcn_wmma_f32_16x16x4_f32(false, a, false, b1,
                                                     (short)0, acc1, false, false);
    }
    int col0 = tn2 * 32 + l16;
    int col1 = col0 + 16;
    float bv0 = bias ? bias[col0] : 0.0f;
    float bv1 = bias ? bias[col1] : 0.0f;
    int row0 = tm * 16 + half * 8;                         // C layout: M split 0-7 / 8-15
#pragma unroll
    for (int i = 0; i < 8; ++i) {
        float v0 = acc0[i] + bv0;
        float v1 = acc1[i] + bv1;
        if (act) { v0 = fmaxf(v0, 0.0f); v1 = fmaxf(v1, 0.0f); }
        float* yr = Y + (size_t)(row0 + i) * Ncol;
        yr[col0] = v0;
        yr[col1] = v1;
    }
}

// ---------- tiny first-layer MLP (Kin = 2 or 3): y = relu(x @ w + b) ----------
__global__ void k_enc_in(const float* __restrict__ x, const float* __restrict__ w,
                         const float* __restrict__ b, float* __restrict__ y,
                         int M, int Kin) {
    int i = blockIdx.x * blockDim.x + threadIdx.x;
    if (i >= M * 64) return;
    int n = i >> 6, j = i & 63;
    float s = b[j];
    for (int kk = 0; kk < Kin; ++kk) s += x[(size_t)n * Kin + kk] * w[kk * 64 + j];
    y[i] = fmaxf(s, 0.0f);
}

// ---------- self-loop attr = mean of incoming edge attrs ----------
__global__ void k_cnt(const int* __restrict__ dst, float* __restrict__ cnt, int E_) {
    int e = blockIdx.x * blockDim.x + threadIdx.x;
    if (e < E_) atomicAdd(&cnt[dst[e]], 1.0f);
}
__global__ void k_eloop_acc(float* __restrict__ e_f, const int* __restrict__ dst, int E_) {
    int i = blockIdx.x * blockDim.x + threadIdx.x;
    if (i >= E_ * 64) return;
    int e = i >> 6, c = i & 63;
    atomicAdd(&e_f[(size_t)(E_ + dst[e]) * 64 + c], e_f[(size_t)e * 64 + c]);
}
__global__ void k_eloop_div(float* __restrict__ e_f, const float* __restrict__ cnt,
                            int N_, int E_) {
    int i = blockIdx.x * blockDim.x + threadIdx.x;
    if (i >= N_ * 64) return;
    int n = i >> 6, c = i & 63;
    e_f[(size_t)(E_ + n) * 64 + c] /= fmaxf(cnt[n], 1.0f);
}

// ---------- attention: logits + segment max ----------
__global__ void k_init_mx(unsigned* __restrict__ mxu, int n) {
    int i = blockIdx.x * blockDim.x + threadIdx.x;
    if (i < n) mxu[i] = f2ord(-INFINITY);
}
__global__ void k_logits(const float* __restrict__ xl, const float* __restrict__ xr,
                         const float* __restrict__ ee, const float* __restrict__ att,
                         const int* __restrict__ src, const int* __restrict__ dst,
                         float* __restrict__ lg, unsigned* __restrict__ mxu,
                         int EN_, int E_) {
    int gw = (blockIdx.x * blockDim.x + threadIdx.x) >> 5;  // warp per edge
    int lane = threadIdx.x & 31;
    if (gw >= EN_) return;
    int s = gw < E_ ? src[gw] : gw - E_;
    int d = gw < E_ ? dst[gw] : gw - E_;
    const float* pxl = xl + (size_t)s * GHC;
    const float* pxr = xr + (size_t)d * GHC;
    const float* pee = ee + (size_t)gw * GHC;
    __builtin_prefetch(pee + 8 * GHC, 0, 0);                // global_prefetch_b8
    float accs[4];
#pragma unroll
    for (int hh = 0; hh < 4; ++hh) {
        float a = 0.0f;
#pragma unroll
        for (int j = 0; j < 2; ++j) {
            int c = hh * 64 + lane + j * 32;
            float m = pxl[c] + pxr[c] + pee[c];
            m = m > 0.0f ? m : 0.2f * m;                    // leaky_relu
            a += m * att[c];
        }
        accs[hh] = a;
    }
#pragma unroll
    for (int hh = 0; hh < 4; ++hh) {
        float v = accs[hh];
        for (int off = 16; off; off >>= 1) v += __shfl_xor(v, off, 32);
        if (lane == 0) {
            lg[(size_t)gw * 4 + hh] = v;
            atomicMax(&mxu[(size_t)d * 4 + hh], f2ord(v));
        }
    }
}

// ---------- exp(logit - max) + segment sum ----------
__global__ void k_expsum(float* __restrict__ lg, const unsigned* __restrict__ mxu,
                         float* __restrict__ ssum, const int* __restrict__ dst,
                         int EN_, int E_) {
    int i = blockIdx.x * blockDim.x + threadIdx.x;
    if (i >= EN_ * 4) return;
    int e = i >> 2, hh = i & 3;
    int d = e < E_ ? dst[e] : e - E_;
    float ex = expf(lg[i] - ord2f(mxu[(size_t)d * 4 + hh]));
    lg[i] = ex;
    atomicAdd(&ssum[(size_t)d * 4 + hh], ex);
}

// ---------- alpha-weighted scatter-add of xl[src] ----------
__global__ void k_scatter(const float* __restrict__ xl, const float* __restrict__ lg,
                          const float* __restrict__ ssum, const int* __restrict__ src,
                          const int* __restrict__ dst, float* __restrict__ oacc,
                          int EN_, int E_) {
    int gw = (blockIdx.x * blockDim.x + threadIdx.x) >> 5;  // warp per edge
    int lane = threadIdx.x & 31;
    if (gw >= EN_) return;
    int s = gw < E_ ? src[gw] : gw - E_;
    int d = gw < E_ ? dst[gw] : gw - E_;
    float al[4];
#pragma unroll
    for (int hh = 0; hh < 4; ++hh)
        al[hh] = lg[(size_t)gw * 4 + hh] / ssum[(size_t)d * 4 + hh];
    const float* pxl = xl + (size_t)s * GHC;
    float* po = oacc + (size_t)d * GHC;
#pragma unroll
    for (int j = 0; j < 8; ++j) {
        int ch = j * 32 + lane;
        atomicAdd(&po[ch], pxl[ch] * al[ch >> 6]);
    }
}

// ---------- residual update: h += relu(mean over heads + bias) ----------
__global__ void k_update(float* __restrict__ h, const float* __restrict__ oacc,
                         const float* __restrict__ bias, int N_) {
    int i = blockIdx.x * blockDim.x + threadIdx.x;
    if (i >= N_ * 64) return;
    int n = i >> 6, c = i & 63;
    const float* po = oacc + (size_t)n * GHC;
    float v = 0.25f * (po[c] + po[64 + c] + po[128 + c] + po[192 + c]) + bias[c];
    h[i] += fmaxf(v, 0.0f);
}

// ---------- edge classifier feature gather [h[src], h[dst], e] ----------
__global__ void k_gather_feat(const float* __restrict__ h, const float* __restrict__ ef,
                              const int* __restrict__ src, const int* __restrict__ dst,
                              float* __restrict__ feat, int E_) {
    int i = blockIdx.x * blockDim.x + threadIdx.x;
    if (i >= E_ * 192) return;
    int e = i / 192, j = i - e * 192;
    float v;
    if (j < 64)       v = h[(size_t)src[e] * 64 + j];
    else if (j < 128) v = h[(size_t)dst[e] * 64 + (j - 64)];
    else              v = ef[(size_t)e * 64 + (j - 128)];
    feat[i] = v;
}

// ---------- final heads ----------
__global__ void k_score(const float* __restrict__ hid, const float* __restrict__ w2,
                        const float* __restrict__ b2, float* __restrict__ out, int E_) {
    int e = blockIdx.x * blockDim.x + threadIdx.x;
    if (e >= E_) return;
    float s = b2[0];
    const float* ph = hid + (size_t)e * 64;
#pragma unroll
    for (int k = 0; k < 64; ++k) s += ph[k] * w2[k];
    out[e] = 1.0f / (1.0f + expf(-s));
}
__global__ void k_offsets(const float* __restrict__ oh, const float* __restrict__ w2,
                          const float* __restrict__ b2, float* __restrict__ out, int N_) {
    int i = blockIdx.x * blockDim.x + threadIdx.x;
    if (i >= N_ * 2) return;
    int n = i >> 1, j = i & 1;
    float s = b2[j];
    const float* ph = oh + (size_t)n * 64;
#pragma unroll
    for (int k = 0; k < 64; ++k) s += ph[k] * w2[k * 2 + j];
    out[i] = s;
}

// ---------- host ----------
template <int K>
static inline void launch_gemm(const float* X, const float* W, const float* bias,
                               float* Y, int M, int Ncol, int act,
                               hipStream_t stream) {
    int waves = (M >> 4) * (Ncol >> 5);   // one wave per 16x32 strip
    int threads = waves * 32;
    hipLaunchKernelGGL((k_gemm_wmma<K>), dim3(CDIV(threads, 256)), dim3(256), 0, stream,
                       X, W, bias, Y, M, Ncol, act);
}

extern "C" void kernel_launch(void* const* d_in, const int* in_sizes, int n_in,
                              void* d_out, int out_size, void* d_ws, size_t ws_size,
                              hipStream_t stream) {
    (void)in_sizes; (void)n_in; (void)out_size; (void)ws_size;
    const float* x     = (const float*)d_in[0];
    const int*   ei    = (const int*)d_in[1];
    const float* ea    = (const float*)d_in[2];
    const float* ne_w1 = (const float*)d_in[3];
    const float* ne_b1 = (const float*)d_in[4];
    const float* ne_w2 = (const float*)d_in[5];
    const float* ne_b2 = (const float*)d_in[6];
    const float* ee_w1 = (const float*)d_in[7];
    const float* ee_b1 = (const float*)d_in[8];
    const float* ee_w2 = (const float*)d_in[9];
    const float* ee_b2 = (const float*)d_in[10];
    const float* cv_wl = (const float*)d_in[11];
    const float* cv_bl = (const float*)d_in[12];
    const float* cv_wr = (const float*)d_in[13];
    const float* cv_br = (const float*)d_in[14];
    const float* cv_we = (const float*)d_in[15];
    const float* cv_att  = (const float*)d_in[16];
    const float* cv_bias = (const float*)d_in[17];
    const float* ec_w1 = (const float*)d_in[18];
    const float* ec_b1 = (const float*)d_in[19];
    const float* ec_w2 = (const float*)d_in[20];
    const float* ec_b2 = (const float*)d_in[21];
    const float* or_w1 = (const float*)d_in[22];
    const float* or_b1 = (const float*)d_in[23];
    const float* or_w2 = (const float*)d_in[24];
    const float* or_b2 = (const float*)d_in[25];

    const int* src = ei;
    const int* dst = ei + GE;

    // workspace layout (floats)
    float* ws = (float*)d_ws;
    size_t off = 0;
    float* h    = ws + off; off += (size_t)GN * 64;
    float* e_f  = ws + off; off += (size_t)GEN * 64;   // rows [0,E) = e, [E,E+N) = loop attrs
    float* xl   = ws + off; off += (size_t)GN * GHC;
    float* xr   = ws + off; off += (size_t)GN * GHC;
    float* ee   = ws + off; off += (size_t)GEN * GHC;
    float* lg   = ws + off; off += (size_t)GEN * 4;
    unsigned* mxu = (unsigned*)(ws + off); off += (size_t)GN * 4;
    float* ssum = ws + off; off += (size_t)GN * 4;
    float* oacc = ws + off; off += (size_t)GN * GHC;
    float* cnt  = ws + off; off += (size_t)GN;
    // reuse after attention layers:
    float* feat = ee;                                   // E*192 <= EN*256
    float* hid  = ee + (size_t)GE * 192;                // E*64 fits in remainder
    float* t1 = xl;                                     // encoder temps
    float* t2 = ee;
    float* oh = xl;                                     // offset-regressor hidden

    float* out_scores  = (float*)d_out;                 // [E]
    float* out_offsets = (float*)d_out + GE;            // [N,2]

    // ---- node encoder: h = relu(x@w1+b1)@w2+b2 ----
    hipLaunchKernelGGL(k_enc_in, dim3(CDIV(GN * 64, 256)), dim3(256), 0, stream,
                       x, ne_w1, ne_b1, t1, GN, 2);
    launch_gemm<64>(t1, ne_w2, ne_b2, h, GN, 64, 0, stream);

    // ---- edge encoder into e_f[0:E) ----
    hipLaunchKernelGGL(k_enc_in, dim3(CDIV(GE * 64, 256)), dim3(256), 0, stream,
                       ea, ee_w1, ee_b1, t2, GE, 3);
    launch_gemm<64>(t2, ee_w2, ee_b2, e_f, GE, 64, 0, stream);

    // ---- self-loop attrs: mean of incoming edge attrs ----
    hipMemsetAsync(e_f + (size_t)GE * 64, 0, (size_t)GN * 64 * sizeof(float), stream);
    hipMemsetAsync(cnt, 0, (size_t)GN * sizeof(float), stream);
    hipLaunchKernelGGL(k_cnt, dim3(CDIV(GE, 256)), dim3(256), 0, stream, dst, cnt, GE);
    hipLaunchKernelGGL(k_eloop_acc, dim3(CDIV(GE * 64, 256)), dim3(256), 0, stream,
                       e_f, dst, GE);
    hipLaunchKernelGGL(k_eloop_div, dim3(CDIV(GN * 64, 256)), dim3(256), 0, stream,
                       e_f, cnt, GN, GE);

    // ---- 3 GATv2 layers ----
    for (int l = 0; l < 3; ++l) {
        const float* wl = cv_wl + (size_t)l * 64 * GHC;
        const float* wr = cv_wr + (size_t)l * 64 * GHC;
        const float* we = cv_we + (size_t)l * 64 * GHC;
        const float* bl = cv_bl + (size_t)l * GHC;
        const float* br = cv_br + (size_t)l * GHC;
        const float* at = cv_att + (size_t)l * GHC;
        const float* bi = cv_bias + (size_t)l * 64;

        launch_gemm<64>(h, wl, bl, xl, GN, GHC, 0, stream);
        launch_gemm<64>(h, wr, br, xr, GN, GHC, 0, stream);
        launch_gemm<64>(e_f, we, nullptr, ee, GEN, GHC, 0, stream);

        hipLaunchKernelGGL(k_init_mx, dim3(CDIV(GN * 4, 256)), dim3(256), 0, stream,
                           mxu, GN * 4);
        hipMemsetAsync(ssum, 0, (size_t)GN * 4 * sizeof(float), stream);
        hipMemsetAsync(oacc, 0, (size_t)GN * GHC * sizeof(float), stream);

        hipLaunchKernelGGL(k_logits, dim3(CDIV(GEN * 32, 256)), dim3(256), 0, stream,
                           xl, xr, ee, at, src, dst, lg, mxu, GEN, GE);
        hipLaunchKernelGGL(k_expsum, dim3(CDIV(GEN * 4, 256)), dim3(256), 0, stream,
                           lg, mxu, ssum, dst, GEN, GE);
        hipLaunchKernelGGL(k_scatter, dim3(CDIV(GEN * 32, 256)), dim3(256), 0, stream,
                           xl, lg, ssum, src, dst, oacc, GEN, GE);
        hipLaunchKernelGGL(k_update, dim3(CDIV(GN * 64, 256)), dim3(256), 0, stream,
                           h, oacc, bi, GN);
    }

    // ---- edge classifier ----
    hipLaunchKernelGGL(k_gather_feat, dim3(CDIV(GE * 192, 256)), dim3(256), 0, stream,
                       h, e_f, src, dst, feat, GE);
    launch_gemm<192>(feat, ec_w1, ec_b1, hid, GE, 64, 1, stream);
    hipLaunchKernelGGL(k_score, dim3(CDIV(GE, 256)), dim3(256), 0, stream,
                       hid, ec_w2, ec_b2, out_scores, GE);

    // ---- offset regressor ----
    launch_gemm<64>(h, or_w1, or_b1, oh, GN, 64, 1, stream);
    hipLaunchKernelGGL(k_offsets, dim3(CDIV(GN * 2, 256)), dim3(256), 0, stream,
                       oh, or_w2, or_b2, out_offsets, GN);
}